// WRF_10763188044350
// MI455X (gfx1250) — compile-verified
//
#include <hip/hip_runtime.h>
#include <math.h>

// ---------------------------------------------------------------------------
// WRF-like RK3 step on a 40x256x256 staggered grid (fp32).
// Memory-bound stencil: ~1.2 GFLOP vs ~0.9 GB of traffic -> HBM/L2 bound.
// Strategy: per RK stage, one "intermediates" kernel (u,v,w,alpha,theta,p,Omega)
// followed by fused update kernels; intermediates stay L2-resident (192MB L2).
// ---------------------------------------------------------------------------

namespace {
constexpr int NZ = 40;
constexpr int NY = 256;
constexpr int NX = 256;
constexpr int NCn = NZ * NY * NX;          // center fields (Theta, Mu, alpha, theta, p)
constexpr int NUn = NZ * NY * (NX + 1);    // U / u
constexpr int NVn = NZ * (NY + 1) * NX;    // V / v
constexpr int NWn = (NZ - 1) * NY * NX;    // W / w / Omega / Phi
}

__device__ __forceinline__ int idxC(int k, int j, int i) { return (k * NY + j) * NX + i; }
__device__ __forceinline__ int idxU(int k, int j, int i) { return (k * NY + j) * (NX + 1) + i; }
__device__ __forceinline__ int idxV(int k, int j, int i) { return (k * (NY + 1) + j) * NX + i; }
__device__ __forceinline__ int idxW(int k, int j, int i) { return (k * NY + j) * NX + i; }

// z accessors with Dirichlet-0 padding (pad_z semantics)
__device__ __forceinline__ float wfld(const float* X, int m, int j, int i) {
  return (m >= 0 && m <= NZ - 2) ? X[idxW(m, j, i)] : 0.f;
}
__device__ __forceinline__ float cfld(const float* X, int m, int j, int i) {
  return (m >= 0 && m <= NZ - 1) ? X[idxC(m, j, i)] : 0.f;
}
__device__ __forceinline__ float ufld(const float* X, int m, int j, int i) {
  return (m >= 0 && m <= NZ - 1) ? X[idxU(m, j, i)] : 0.f;
}
__device__ __forceinline__ float vfld(const float* X, int m, int j, int i) {
  return (m >= 0 && m <= NZ - 1) ? X[idxV(m, j, i)] : 0.f;
}
// bar_z(pad_z(p))[kp], kp in [0, NZ]
__device__ __forceinline__ float pzf(const float* p, int kp, int j, int i) {
  return 0.5f * (cfld(p, kp - 1, j, i) + cfld(p, kp, j, i));
}
// H = bar_z(pad_z(Phi))[m], m in [0, NZ-1]
__device__ __forceinline__ float Hphi(const float* Phi, int m, int j, int i) {
  return 0.5f * (wfld(Phi, m - 1, j, i) + wfld(Phi, m, j, i));
}

// ---------------------------------------------------------------------------
// Intermediates: u, v, w, alpha, theta, p, Omega from the stage state.
// ---------------------------------------------------------------------------
__global__ __launch_bounds__(256) void k_inter(
    const float* __restrict__ U, const float* __restrict__ V,
    const float* __restrict__ W, const float* __restrict__ Theta,
    const float* __restrict__ Mu, const float* __restrict__ Phi,
    float* __restrict__ u, float* __restrict__ v, float* __restrict__ w,
    float* __restrict__ alpha, float* __restrict__ theta,
    float* __restrict__ p, float* __restrict__ Om,
    const float* __restrict__ pdz, const float* __restrict__ pPREF,
    const float* __restrict__ pRd, const float* __restrict__ pg) {
  int t = blockIdx.x * blockDim.x + threadIdx.x;
  if (t >= NCn) return;
  int i = t & (NX - 1);
  int j = (t >> 8) & (NY - 1);
  int k = t >> 16;
  const float dz = *pdz, PREF = *pPREF, Rd = *pRd, g = *pg;

  // gfx1250 prefetch path: pull next z-plane line into cache (global_prefetch_b8)
  if (k + 1 < NZ) __builtin_prefetch(&Mu[idxC(k + 1, j, i)], 0, 3);

  const float mu_c = Mu[idxC(k, j, i)];
  const float phim = (k >= 1) ? Phi[idxW(k - 1, j, i)] : 0.f;
  const float phik = (k <= NZ - 2) ? Phi[idxW(k, j, i)] : 0.f;
  const float al = -((phik - phim) / dz) / mu_c;
  const float th = Theta[idxC(k, j, i)] / mu_c;
  const float pp = PREF * powf((Rd * th / PREF) / al, 1.4f);
  alpha[idxC(k, j, i)] = al;
  theta[idxC(k, j, i)] = th;
  p[idxC(k, j, i)] = pp;

  // u = U / bar_x(pad_x(Mu))   (periodic x)
  const float mux = 0.5f * (Mu[idxC(k, j, (i - 1) & (NX - 1))] + mu_c);
  u[idxU(k, j, i)] = U[idxU(k, j, i)] / mux;
  if (i == 0) {  // extra staggered column i == NX
    const float muxE = 0.5f * (Mu[idxC(k, j, NX - 1)] + Mu[idxC(k, j, 0)]);
    u[idxU(k, j, NX)] = U[idxU(k, j, NX)] / muxE;
  }
  // v = V / bar_y(pad_y(Mu))   (periodic y)
  const float muy = 0.5f * (Mu[idxC(k, (j - 1) & (NY - 1), i)] + mu_c);
  v[idxV(k, j, i)] = V[idxV(k, j, i)] / muy;
  if (j == 0) {  // extra staggered row j == NY
    const float muyE = 0.5f * (Mu[idxC(k, NY - 1, i)] + Mu[idxC(k, 0, i)]);
    v[idxV(k, NY, i)] = V[idxV(k, NY, i)] / muyE;
  }
  // w = W / bar_z(Mu);  Omega = -w*g / bar_z(alpha)
  if (k < NZ - 1) {
    const float mu_p = Mu[idxC(k + 1, j, i)];
    const float ww = W[idxW(k, j, i)] / (0.5f * (mu_c + mu_p));
    const float phitop = (k + 1 <= NZ - 2) ? Phi[idxW(k + 1, j, i)] : 0.f;
    const float alp = -((phitop - phik) / dz) / mu_p;
    w[idxW(k, j, i)] = ww;
    Om[idxW(k, j, i)] = -ww * g / (0.5f * (al + alp));
  }
}

// ---------------------------------------------------------------------------
// dst_U = base_U + c * R_U     (shape NZ x NY x (NX+1))
// ---------------------------------------------------------------------------
__global__ __launch_bounds__(256) void k_RU(
    const float* __restrict__ baseU,
    const float* __restrict__ U, const float* __restrict__ V,
    const float* __restrict__ Mu, const float* __restrict__ Phi,
    const float* __restrict__ u, const float* __restrict__ alpha,
    const float* __restrict__ p, const float* __restrict__ Om,
    float* __restrict__ dst,
    const float* __restrict__ pdx, const float* __restrict__ pdy,
    const float* __restrict__ pdz, const float* __restrict__ pdt,
    float stageDiv) {
  int t = blockIdx.x * blockDim.x + threadIdx.x;
  if (t >= NUn) return;
  const int i = t % (NX + 1);
  const int jk = t / (NX + 1);
  const int j = jk & (NY - 1);
  const int k = jk >> 8;
  const float dx = *pdx, dy = *pdy, dz = *pdz;
  const float c = (*pdt) / stageDiv;

  // term1: -dlt_x( bar_x(pad_x(U)) * bar_x(pad_x(u)) )   (wrap over NX+1 staggered cols)
  const int xm = (i == 0) ? NX : (i - 1);
  const int xp = (i == NX) ? 0 : (i + 1);
  const float Uc = U[idxU(k, j, i)], Um = U[idxU(k, j, xm)], Up = U[idxU(k, j, xp)];
  const float uc = u[idxU(k, j, i)], um = u[idxU(k, j, xm)], up = u[idxU(k, j, xp)];
  const float term1 =
      -((0.5f * (Uc + Up)) * (0.5f * (uc + up)) - (0.5f * (Um + Uc)) * (0.5f * (um + uc))) / dx;

  const int icm = (i - 1) & (NX - 1);
  const int ic0 = i & (NX - 1);
  const int jm = (j - 1) & (NY - 1);
  const int jpw = (j + 1) & (NY - 1);

  // term2: -dlt_y( bar_x(pad_x(V)) * bar_y(pad_y(u)) )
  const float Cj  = 0.5f * (V[idxV(k, j,     icm)] + V[idxV(k, j,     ic0)]);
  const float Cj1 = 0.5f * (V[idxV(k, j + 1, icm)] + V[idxV(k, j + 1, ic0)]);
  const float Dj  = 0.5f * (u[idxU(k, jm, i)] + uc);
  const float Dj1 = 0.5f * (uc + u[idxU(k, jpw, i)]);
  const float term2 = -(Cj1 * Dj1 - Cj * Dj) / dy;

  // term3: -dlt_z( bar_x(pad_x(pad_z(Omega))) * bar_z(pad_z(u)) )
  const float Ek  = 0.5f * (wfld(Om, k - 1, j, icm) + wfld(Om, k - 1, j, ic0));
  const float Ek1 = 0.5f * (wfld(Om, k,     j, icm) + wfld(Om, k,     j, ic0));
  const float Fk  = 0.5f * (ufld(u, k - 1, j, i) + uc);
  const float Fk1 = 0.5f * (uc + ufld(u, k + 1, j, i));
  const float term3 = -(Ek1 * Fk1 - Ek * Fk) / dz;

  // pressure terms
  const float mux = 0.5f * (Mu[idxC(k, j, icm)] + Mu[idxC(k, j, ic0)]);
  const float alx = 0.5f * (alpha[idxC(k, j, icm)] + alpha[idxC(k, j, ic0)]);
  const float dpx = (p[idxC(k, j, ic0)] - p[idxC(k, j, icm)]) / dx;
  const float Gk  = 0.5f * (pzf(p, k,     j, icm) + pzf(p, k,     j, ic0));
  const float Gk1 = 0.5f * (pzf(p, k + 1, j, icm) + pzf(p, k + 1, j, ic0));
  const float dGz = (Gk1 - Gk) / dz;
  const float dHx = (Hphi(Phi, k, j, ic0) - Hphi(Phi, k, j, icm)) / dx;

  const float R = term1 + term2 + term3 - mux * alx * dpx - dGz * dHx;
  dst[idxU(k, j, i)] = baseU[idxU(k, j, i)] + c * R;
}

// ---------------------------------------------------------------------------
// dst_V = base_V + c * R_V     (shape NZ x (NY+1) x NX)
// ---------------------------------------------------------------------------
__global__ __launch_bounds__(256) void k_RV(
    const float* __restrict__ baseV,
    const float* __restrict__ U, const float* __restrict__ V,
    const float* __restrict__ Mu, const float* __restrict__ Phi,
    const float* __restrict__ v, const float* __restrict__ alpha,
    const float* __restrict__ p, const float* __restrict__ Om,
    float* __restrict__ dst,
    const float* __restrict__ pdx, const float* __restrict__ pdy,
    const float* __restrict__ pdz, const float* __restrict__ pdt,
    float stageDiv) {
  int t = blockIdx.x * blockDim.x + threadIdx.x;
  if (t >= NVn) return;
  const int i = t & (NX - 1);
  const int jk = t >> 8;
  const int j = jk % (NY + 1);
  const int k = jk / (NY + 1);
  const float dx = *pdx, dy = *pdy, dz = *pdz;
  const float c = (*pdt) / stageDiv;

  const int im = (i - 1) & (NX - 1);
  const int ipw = (i + 1) & (NX - 1);
  const int jmw = (j - 1) & (NY - 1);
  const int j0w = j & (NY - 1);
  const float vc = v[idxV(k, j, i)];

  // term1: -dlt_x( bar_y(pad_y(U)) * bar_x(pad_x(v)) )
  const float Ubi  = 0.5f * (U[idxU(k, jmw, i)]     + U[idxU(k, j0w, i)]);
  const float Ubi1 = 0.5f * (U[idxU(k, jmw, i + 1)] + U[idxU(k, j0w, i + 1)]);
  const float vbi  = 0.5f * (v[idxV(k, j, im)] + vc);
  const float vbi1 = 0.5f * (vc + v[idxV(k, j, ipw)]);
  const float term1 = -(Ubi1 * vbi1 - Ubi * vbi) / dx;

  // term2: -dlt_y( bar_y(pad_y(V)) * bar_y(pad_y(v)) )   (wrap over NY+1 staggered rows)
  const int ym = (j == 0) ? NY : (j - 1);
  const int yp = (j == NY) ? 0 : (j + 1);
  const float Vc = V[idxV(k, j, i)], Vm = V[idxV(k, ym, i)], Vp = V[idxV(k, yp, i)];
  const float vm = v[idxV(k, ym, i)], vp = v[idxV(k, yp, i)];
  const float term2 =
      -((0.5f * (Vc + Vp)) * (0.5f * (vc + vp)) - (0.5f * (Vm + Vc)) * (0.5f * (vm + vc))) / dy;

  // term3: -dlt_z( bar_y(pad_y(pad_z(Omega))) * bar_z(pad_z(v)) )
  const float Ek  = 0.5f * (wfld(Om, k - 1, jmw, i) + wfld(Om, k - 1, j0w, i));
  const float Ek1 = 0.5f * (wfld(Om, k,     jmw, i) + wfld(Om, k,     j0w, i));
  const float Fk  = 0.5f * (vfld(v, k - 1, j, i) + vc);
  const float Fk1 = 0.5f * (vc + vfld(v, k + 1, j, i));
  const float term3 = -(Ek1 * Fk1 - Ek * Fk) / dz;

  // pressure terms
  const float muy = 0.5f * (Mu[idxC(k, jmw, i)] + Mu[idxC(k, j0w, i)]);
  const float aly = 0.5f * (alpha[idxC(k, jmw, i)] + alpha[idxC(k, j0w, i)]);
  const float dpy = (p[idxC(k, j0w, i)] - p[idxC(k, jmw, i)]) / dy;
  const float Gk  = 0.5f * (pzf(p, k,     jmw, i) + pzf(p, k,     j0w, i));
  const float Gk1 = 0.5f * (pzf(p, k + 1, jmw, i) + pzf(p, k + 1, j0w, i));
  const float dGz = (Gk1 - Gk) / dz;
  const float dHy = (Hphi(Phi, k, j0w, i) - Hphi(Phi, k, jmw, i)) / dy;

  const float R = term1 + term2 + term3 - muy * aly * dpy - dGz * dHy;
  dst[idxV(k, j, i)] = baseV[idxV(k, j, i)] + c * R;
}

// ---------------------------------------------------------------------------
// dst_W = base_W + c * R_W     (shape (NZ-1) x NY x NX)
// ---------------------------------------------------------------------------
__global__ __launch_bounds__(256) void k_RW(
    const float* __restrict__ baseW,
    const float* __restrict__ U, const float* __restrict__ V,
    const float* __restrict__ Mu, const float* __restrict__ p,
    const float* __restrict__ w, const float* __restrict__ Om,
    float* __restrict__ dst,
    const float* __restrict__ pdx, const float* __restrict__ pdy,
    const float* __restrict__ pdz, const float* __restrict__ pdt,
    const float* __restrict__ pg, float stageDiv) {
  int t = blockIdx.x * blockDim.x + threadIdx.x;
  if (t >= NWn) return;
  const int i = t & (NX - 1);
  const int j = (t >> 8) & (NY - 1);
  const int k = t >> 16;  // 0 .. NZ-2
  const float dx = *pdx, dy = *pdy, dz = *pdz, g = *pg;
  const float c = (*pdt) / stageDiv;

  const int im = (i - 1) & (NX - 1);
  const int ipw = (i + 1) & (NX - 1);
  const int jm = (j - 1) & (NY - 1);
  const int jpw = (j + 1) & (NY - 1);
  const float wc = w[idxW(k, j, i)];

  const float Uzi  = 0.5f * (U[idxU(k, j, i)]     + U[idxU(k + 1, j, i)]);
  const float Uzi1 = 0.5f * (U[idxU(k, j, i + 1)] + U[idxU(k + 1, j, i + 1)]);
  const float wbi  = 0.5f * (w[idxW(k, j, im)] + wc);
  const float wbi1 = 0.5f * (wc + w[idxW(k, j, ipw)]);
  const float term1 = -(Uzi1 * wbi1 - Uzi * wbi) / dx;

  const float Vzj  = 0.5f * (V[idxV(k, j, i)]     + V[idxV(k + 1, j, i)]);
  const float Vzj1 = 0.5f * (V[idxV(k, j + 1, i)] + V[idxV(k + 1, j + 1, i)]);
  const float wbj  = 0.5f * (w[idxW(k, jm, i)] + wc);
  const float wbj1 = 0.5f * (wc + w[idxW(k, jpw, i)]);
  const float term2 = -(Vzj1 * wbj1 - Vzj * wbj) / dy;

  const float Omc = Om[idxW(k, j, i)];
  const float Ozk  = 0.5f * (wfld(Om, k - 1, j, i) + Omc);
  const float Ozk1 = 0.5f * (Omc + wfld(Om, k + 1, j, i));
  const float Wzk  = 0.5f * (wfld(w, k - 1, j, i) + wc);
  const float Wzk1 = 0.5f * (wc + wfld(w, k + 1, j, i));
  const float term3 = -(Ozk1 * Wzk1 - Ozk * Wzk) / dz;

  const float addw = g * ((p[idxC(k + 1, j, i)] - p[idxC(k, j, i)]) / dz -
                          0.5f * (Mu[idxC(k, j, i)] + Mu[idxC(k + 1, j, i)]));

  dst[idxW(k, j, i)] = baseW[idxW(k, j, i)] + c * (term1 + term2 + term3 + addw);
}

// ---------------------------------------------------------------------------
// Fused Theta/Mu update     (shape NZ x NY x NX)
// ---------------------------------------------------------------------------
__global__ __launch_bounds__(256) void k_RTM(
    const float* __restrict__ baseT, const float* __restrict__ baseM,
    const float* __restrict__ U, const float* __restrict__ V,
    const float* __restrict__ theta, const float* __restrict__ Om,
    float* __restrict__ dstT, float* __restrict__ dstM,
    const float* __restrict__ pdx, const float* __restrict__ pdy,
    const float* __restrict__ pdz, const float* __restrict__ pdt,
    float stageDiv) {
  int t = blockIdx.x * blockDim.x + threadIdx.x;
  if (t >= NCn) return;
  const int i = t & (NX - 1);
  const int j = (t >> 8) & (NY - 1);
  const int k = t >> 16;
  const float dx = *pdx, dy = *pdy, dz = *pdz;
  const float c = (*pdt) / stageDiv;

  const int im = (i - 1) & (NX - 1);
  const int ipw = (i + 1) & (NX - 1);
  const int jm = (j - 1) & (NY - 1);
  const int jpw = (j + 1) & (NY - 1);

  const float Ui = U[idxU(k, j, i)], Ui1 = U[idxU(k, j, i + 1)];
  const float Vj = V[idxV(k, j, i)], Vj1 = V[idxV(k, j + 1, i)];
  const float th_c = theta[idxC(k, j, i)];

  const float fx0 = Ui  * 0.5f * (theta[idxC(k, j, im)] + th_c);
  const float fx1 = Ui1 * 0.5f * (th_c + theta[idxC(k, j, ipw)]);
  const float fy0 = Vj  * 0.5f * (theta[idxC(k, jm, i)] + th_c);
  const float fy1 = Vj1 * 0.5f * (th_c + theta[idxC(k, jpw, i)]);

  const float Omk  = wfld(Om, k - 1, j, i);   // Omegaz[k]
  const float Omk1 = wfld(Om, k,     j, i);   // Omegaz[k+1]
  const float Thzk  = 0.5f * (cfld(theta, k - 1, j, i) + th_c);
  const float Thzk1 = 0.5f * (th_c + cfld(theta, k + 1, j, i));

  const float RT = -(fx1 - fx0) / dx - (fy1 - fy0) / dy - (Omk1 * Thzk1 - Omk * Thzk) / dz;
  const float RM = -(Ui1 - Ui) / dx - (Vj1 - Vj) / dy - (Omk1 - Omk) / dz;

  dstT[idxC(k, j, i)] = baseT[idxC(k, j, i)] + c * RT;
  dstM[idxC(k, j, i)] = baseM[idxC(k, j, i)] + c * RM;
}

// ---------------------------------------------------------------------------
// dst_Phi = base_Phi + c * R_Phi     (shape (NZ-1) x NY x NX)
// ---------------------------------------------------------------------------
__global__ __launch_bounds__(256) void k_RPhi(
    const float* __restrict__ basePhi, const float* __restrict__ Phi,
    const float* __restrict__ u, const float* __restrict__ v,
    const float* __restrict__ w, const float* __restrict__ Om,
    float* __restrict__ dst,
    const float* __restrict__ pdx, const float* __restrict__ pdy,
    const float* __restrict__ pdz, const float* __restrict__ pdt,
    const float* __restrict__ pg, float stageDiv) {
  int t = blockIdx.x * blockDim.x + threadIdx.x;
  if (t >= NWn) return;
  const int i = t & (NX - 1);
  const int j = (t >> 8) & (NY - 1);
  const int k = t >> 16;  // 0 .. NZ-2
  const float dx = *pdx, dy = *pdy, dz = *pdz, g = *pg;
  const float c = (*pdt) / stageDiv;

  const int im = (i - 1) & (NX - 1);
  const int ipw = (i + 1) & (NX - 1);
  const int jm = (j - 1) & (NY - 1);
  const int jpw = (j + 1) & (NY - 1);

  const float ubz = 0.25f * (u[idxU(k, j, i)] + u[idxU(k, j, i + 1)] +
                             u[idxU(k + 1, j, i)] + u[idxU(k + 1, j, i + 1)]);
  const float vbz = 0.25f * (v[idxV(k, j, i)] + v[idxV(k, j + 1, i)] +
                             v[idxV(k + 1, j, i)] + v[idxV(k + 1, j + 1, i)]);

  const float Phic = Phi[idxW(k, j, i)];
  const float dPhix =
      (0.5f * (Phic + Phi[idxW(k, j, ipw)]) - 0.5f * (Phi[idxW(k, j, im)] + Phic)) / dx;
  const float dPhiy =
      (0.5f * (Phic + Phi[idxW(k, jpw, i)]) - 0.5f * (Phi[idxW(k, jm, i)] + Phic)) / dy;
  const float dHz = (Hphi(Phi, k + 1, j, i) - Hphi(Phi, k, j, i)) / dz;

  const float R = -ubz * dPhix - vbz * dPhiy - Om[idxW(k, j, i)] * dHz + g * w[idxW(k, j, i)];
  dst[idxW(k, j, i)] = basePhi[idxW(k, j, i)] + c * R;
}

// ---------------------------------------------------------------------------
// Host side
// ---------------------------------------------------------------------------
extern "C" void kernel_launch(void* const* d_in, const int* in_sizes, int n_in,
                              void* d_out, int out_size, void* d_ws, size_t ws_size,
                              hipStream_t stream) {
  (void)in_sizes; (void)n_in; (void)out_size;

  const float* inU = (const float*)d_in[0];
  const float* inV = (const float*)d_in[1];
  const float* inW = (const float*)d_in[2];
  const float* inT = (const float*)d_in[3];
  const float* inM = (const float*)d_in[4];
  const float* inP = (const float*)d_in[5];
  // d_in[6..11] (Phi_t .. P_s) are ignored, matching the reference.
  const float* pdx   = (const float*)d_in[12];
  const float* pdy   = (const float*)d_in[13];
  const float* pdz   = (const float*)d_in[14];
  const float* pdt   = (const float*)d_in[15];
  const float* pPREF = (const float*)d_in[16];
  const float* pRd   = (const float*)d_in[17];
  const float* pg    = (const float*)d_in[18];

  // Workspace carve-up
  float* ws = (float*)d_ws;
  float* bu  = ws; ws += NUn;   // u
  float* bv  = ws; ws += NVn;   // v
  float* bw  = ws; ws += NWn;   // w
  float* bal = ws; ws += NCn;   // alpha
  float* bth = ws; ws += NCn;   // theta
  float* bp  = ws; ws += NCn;   // p
  float* bom = ws; ws += NWn;   // Omega

  const size_t stateFloats =
      (size_t)NUn + NVn + NWn + NCn + NCn + NWn;   // [U|V|W|Theta|Mu|Phi]
  const size_t offU = 0;
  const size_t offV = offU + NUn;
  const size_t offW = offV + NVn;
  const size_t offT = offW + NWn;
  const size_t offM = offT + NCn;
  const size_t offP = offM + NCn;

  float* Sa = ws; ws += stateFloats;
  const size_t usedOne = (size_t)(ws - (float*)d_ws) * sizeof(float);
  const bool twoBuf = ws_size >= usedOne + stateFloats * sizeof(float);
  float* Sb = twoBuf ? ws : nullptr;
  float* Out = (float*)d_out;

  const dim3 blk(256);
  const int gC = (NCn + 255) / 256;
  const int gU = (NUn + 255) / 256;
  const int gV = (NVn + 255) / 256;
  const int gW = (NWn + 255) / 256;

  // Stage-state starts as the original inputs.
  const float *sU = inU, *sV = inV, *sW = inW, *sT = inT, *sM = inM, *sP = inP;

  for (int s = 0; s < 3; ++s) {
    const float div = (s == 0) ? 3.f : (s == 1) ? 2.f : 1.f;

    float* D;
    if (twoBuf) {
      D = (s == 0) ? Sa : (s == 1) ? Sb : Out;
    } else {
      // Sa -> d_out -> Sa, then one D2D copy at the end.
      D = (s == 0) ? Sa : (s == 1) ? Out : Sa;
    }
    float* dU = D + offU; float* dV = D + offV; float* dW = D + offW;
    float* dT = D + offT; float* dM = D + offM; float* dP = D + offP;

    k_inter<<<gC, blk, 0, stream>>>(sU, sV, sW, sT, sM, sP,
                                    bu, bv, bw, bal, bth, bp, bom,
                                    pdz, pPREF, pRd, pg);
    k_RU<<<gU, blk, 0, stream>>>(inU, sU, sV, sM, sP, bu, bal, bp, bom,
                                 dU, pdx, pdy, pdz, pdt, div);
    k_RV<<<gV, blk, 0, stream>>>(inV, sU, sV, sM, sP, bv, bal, bp, bom,
                                 dV, pdx, pdy, pdz, pdt, div);
    k_RW<<<gW, blk, 0, stream>>>(inW, sU, sV, sM, bp, bw, bom,
                                 dW, pdx, pdy, pdz, pdt, pg, div);
    k_RTM<<<gC, blk, 0, stream>>>(inT, inM, sU, sV, bth, bom,
                                  dT, dM, pdx, pdy, pdz, pdt, div);
    k_RPhi<<<gW, blk, 0, stream>>>(inP, sP, bu, bv, bw, bom,
                                   dP, pdx, pdy, pdz, pdt, pg, div);

    sU = dU; sV = dV; sW = dW; sT = dT; sM = dM; sP = dP;
  }

  if (!twoBuf) {
    // Final state landed in Sa; move it to d_out (graph-capture safe).
    hipMemcpyAsync(d_out, Sa, stateFloats * sizeof(float),
                   hipMemcpyDeviceToDevice, stream);
  }
}